// GCN_15204184228224
// MI455X (gfx1250) — compile-verified
//
#include <hip/hip_runtime.h>
#include <cstdint>
#include <cstddef>

#define N_NODES 500000
#define N_EDGES 8000000

typedef __attribute__((ext_vector_type(2))) float v2f;
typedef __attribute__((ext_vector_type(8))) float v8f;

// ---------------------------------------------------------------------------
// deg[i] = 1 (self-loop)
__global__ __launch_bounds__(256) void init_deg_kernel(int* __restrict__ deg, int n) {
    int i = blockIdx.x * blockDim.x + threadIdx.x;
    if (i < n) deg[i] = 1;
}

// Convert int64 edge list -> int32, count in-degree on dst (exact int atomics).
__global__ __launch_bounds__(256) void prep_edges_kernel(const long long* __restrict__ ei,
                                                         int* __restrict__ src,
                                                         int* __restrict__ dst,
                                                         int* __restrict__ deg, int nE) {
    int e = blockIdx.x * blockDim.x + threadIdx.x;
    if (e >= nE) return;
    int s = (int)ei[e];
    int d = (int)ei[(size_t)nE + e];
    src[e] = s;
    dst[e] = d;
    atomicAdd(&deg[d], 1);
}

// dinv[i] = rsqrt(deg) (deg >= 1 always due to self-loop) — in-place alias of deg.
__global__ __launch_bounds__(256) void dinv_kernel(const int* __restrict__ deg,
                                                   float* __restrict__ dinv, int n) {
    int i = blockIdx.x * blockDim.x + threadIdx.x;
    if (i < n) {
        float d = (float)deg[i];
        dinv[i] = rsqrtf(d);
    }
}

// ---------------------------------------------------------------------------
// Layer-1 transform via V_WMMA_F32_16X16X4_F32:
//   g[i][c] = dinv[i] * sum_k x[i][k] * W1[c][k],  c = 0..3, K = 128 (32 WMMAs)
// One wave per 16-node tile. N_NODES % 16 == 0, so EXEC stays all-ones.
// A (16x4 f32) lane layout: lane L -> row M = L%16, VGPR v -> K = 2*(L/16)+v.
// B (4x16 f32) lane layout: lane L -> col N = L%16, VGPR v -> K = 2*(L/16)+v.
// D (16x16 f32): lane L, VGPR v -> row = v + 8*(L/16), col = L%16.
__global__ __launch_bounds__(128) void transform1_wmma_kernel(
        const float* __restrict__ x, const float* __restrict__ W1,
        const float* __restrict__ dinv, float* __restrict__ g, int nTiles) {
    int wave = blockIdx.x * (blockDim.x >> 5) + (threadIdx.x >> 5);
    int lane = threadIdx.x & 31;
    if (wave >= nTiles) return;              // wave-uniform exit

    int half = lane >> 4;                    // 0 or 1
    int idx  = lane & 15;                    // A row / B-D column
    size_t node = (size_t)wave * 16 + idx;

    const float* xrow = x + node * 128 + 2 * half;
    float bmask = (idx < 4) ? 1.0f : 0.0f;   // only 4 valid output channels
    const float* wrow = W1 + (size_t)((idx < 4) ? idx : 0) * 128 + 2 * half;

    v8f acc;
#pragma unroll
    for (int i = 0; i < 8; ++i) acc[i] = 0.0f;

#pragma unroll 8
    for (int kc = 0; kc < 32; ++kc) {        // K = 128 in chunks of 4
        float2 av = *(const float2*)(xrow + kc * 4);
        float2 bv = *(const float2*)(wrow + kc * 4);
        v2f A; A.x = av.x;         A.y = av.y;
        v2f B; B.x = bv.x * bmask; B.y = bv.y * bmask;
        acc = __builtin_amdgcn_wmma_f32_16x16x4_f32(
                  false, A, false, B, (short)0, acc, false, false);
    }

    if (idx < 4) {                           // columns 0..3 hold real outputs
        int rbase = wave * 16 + 8 * half;
#pragma unroll
        for (int v = 0; v < 8; ++v) {
            int r = rbase + v;
            g[(size_t)r * 4 + idx] = acc[v] * dinv[r];
        }
    }
}

// ---------------------------------------------------------------------------
// Skinny transforms (Cin=4): g[i][c] = dinv[i] * sum_k h[i][k]*W[c][k]
__global__ __launch_bounds__(256) void transform_small_kernel(
        const float* __restrict__ h, const float* __restrict__ W,
        const float* __restrict__ dinv, float* __restrict__ g,
        int n, int cout) {
    int i = blockIdx.x * blockDim.x + threadIdx.x;
    if (i >= n) return;
    float4 hv = *(const float4*)(h + (size_t)i * 4);
    float dv = dinv[i];
    for (int c = 0; c < cout; ++c) {
        const float* w = W + c * 4;
        float s = hv.x * w[0] + hv.y * w[1] + hv.z * w[2] + hv.w * w[3];
        g[(size_t)i * cout + c] = dv * s;
    }
}

__global__ __launch_bounds__(256) void zero_kernel(float* __restrict__ p, int n) {
    int i = blockIdx.x * blockDim.x + threadIdx.x;
    if (i < n) p[i] = 0.0f;
}

// acc[dst] += g[src]  (messages; the dinv factors are pre/post applied)
template <int COUT>
__global__ __launch_bounds__(256) void scatter_kernel(
        const int* __restrict__ src, const int* __restrict__ dst,
        const float* __restrict__ g, float* __restrict__ acc, int nE) {
    int e = blockIdx.x * blockDim.x + threadIdx.x;
    if (e >= nE) return;
    int s = src[e], d = dst[e];
    if (COUT == 4) {
        float4 v = *(const float4*)(g + (size_t)s * 4);
        float* a = acc + (size_t)d * 4;
        unsafeAtomicAdd(a + 0, v.x);
        unsafeAtomicAdd(a + 1, v.y);
        unsafeAtomicAdd(a + 2, v.z);
        unsafeAtomicAdd(a + 3, v.w);
    } else {
        float2 v = *(const float2*)(g + (size_t)s * 2);
        float* a = acc + (size_t)d * 2;
        unsafeAtomicAdd(a + 0, v.x);
        unsafeAtomicAdd(a + 1, v.y);
    }
}

// h_out[i][c] = tanh( dinv[i]*(acc[i][c] + g[i][c]) + b[c] )   (g[i] is self-loop term)
__global__ __launch_bounds__(256) void finalize_kernel(
        const float* __restrict__ acc, const float* __restrict__ g,
        const float* __restrict__ dinv, const float* __restrict__ b,
        float* __restrict__ h, int n, int cout) {
    int i = blockIdx.x * blockDim.x + threadIdx.x;
    if (i >= n) return;
    float dv = dinv[i];
    for (int c = 0; c < cout; ++c) {
        size_t o = (size_t)i * cout + c;
        h[o] = tanhf(dv * (acc[o] + g[o]) + b[c]);
    }
}

// logits = h3 @ Wc^T + bc ; also emit h3 (outputs concatenated flat)
__global__ __launch_bounds__(256) void head_kernel(
        const float* __restrict__ h, const float* __restrict__ Wc,
        const float* __restrict__ bc, float* __restrict__ out, int n) {
    int i = blockIdx.x * blockDim.x + threadIdx.x;
    if (i >= n) return;
    float2 hv = *(const float2*)(h + (size_t)i * 2);
    float* lo = out + (size_t)i * 10;
    for (int j = 0; j < 10; ++j)
        lo[j] = hv.x * Wc[j * 2 + 0] + hv.y * Wc[j * 2 + 1] + bc[j];
    float* hcopy = out + (size_t)n * 10 + (size_t)i * 2;
    hcopy[0] = hv.x;
    hcopy[1] = hv.y;
}

// ---------------------------------------------------------------------------
extern "C" void kernel_launch(void* const* d_in, const int* in_sizes, int n_in,
                              void* d_out, int out_size, void* d_ws, size_t ws_size,
                              hipStream_t stream) {
    (void)in_sizes; (void)n_in; (void)out_size; (void)ws_size;
    const float*     x  = (const float*)d_in[0];
    const long long* ei = (const long long*)d_in[1];
    const float* W1 = (const float*)d_in[2];
    const float* b1 = (const float*)d_in[3];
    const float* W2 = (const float*)d_in[4];
    const float* b2 = (const float*)d_in[5];
    const float* W3 = (const float*)d_in[6];
    const float* b3 = (const float*)d_in[7];
    const float* Wc = (const float*)d_in[8];
    const float* bc = (const float*)d_in[9];
    float* out = (float*)d_out;

    // Workspace layout (all 16B aligned): 90 MB total
    char* ws = (char*)d_ws;
    int*   deg  = (int*)ws;                                   // N ints
    float* dinv = (float*)ws;                                 // alias (after dinv_kernel)
    float* g    = (float*)(ws + (size_t)N_NODES * 4);         // N*4 floats
    float* acc  = g + (size_t)N_NODES * 4;                    // N*4 floats
    float* h    = acc + (size_t)N_NODES * 4;                  // N*4 floats
    int*   src  = (int*)(h + (size_t)N_NODES * 4);            // E ints
    int*   dst  = src + N_EDGES;                              // E ints

    const int TB = 256;
    int gN = (N_NODES + TB - 1) / TB;
    int gE = (N_EDGES + TB - 1) / TB;
    int g4 = (N_NODES * 4 + TB - 1) / TB;
    int g2 = (N_NODES * 2 + TB - 1) / TB;

    // degrees + int32 edges + dinv
    init_deg_kernel<<<gN, TB, 0, stream>>>(deg, N_NODES);
    prep_edges_kernel<<<gE, TB, 0, stream>>>(ei, src, dst, deg, N_EDGES);
    dinv_kernel<<<gN, TB, 0, stream>>>(deg, dinv, N_NODES);

    // Layer 1: WMMA transform (K=128), then gather/scatter aggregate
    int nTiles = N_NODES / 16;                         // 31250, exact
    int gT = (nTiles + 3) / 4;                         // 4 waves / 128-thread block
    transform1_wmma_kernel<<<gT, 128, 0, stream>>>(x, W1, dinv, g, nTiles);
    zero_kernel<<<g4, TB, 0, stream>>>(acc, N_NODES * 4);
    scatter_kernel<4><<<gE, TB, 0, stream>>>(src, dst, g, acc, N_EDGES);
    finalize_kernel<<<gN, TB, 0, stream>>>(acc, g, dinv, b1, h, N_NODES, 4);

    // Layer 2 (4 -> 4)
    transform_small_kernel<<<gN, TB, 0, stream>>>(h, W2, dinv, g, N_NODES, 4);
    zero_kernel<<<g4, TB, 0, stream>>>(acc, N_NODES * 4);
    scatter_kernel<4><<<gE, TB, 0, stream>>>(src, dst, g, acc, N_EDGES);
    finalize_kernel<<<gN, TB, 0, stream>>>(acc, g, dinv, b2, h, N_NODES, 4);

    // Layer 3 (4 -> 2)
    transform_small_kernel<<<gN, TB, 0, stream>>>(h, W3, dinv, g, N_NODES, 2);
    zero_kernel<<<g2, TB, 0, stream>>>(acc, N_NODES * 2);
    scatter_kernel<2><<<gE, TB, 0, stream>>>(src, dst, g, acc, N_EDGES);
    finalize_kernel<<<gN, TB, 0, stream>>>(acc, g, dinv, b3, h, N_NODES, 2);

    // Classifier head + emit h3
    head_kernel<<<gN, TB, 0, stream>>>(h, Wc, bc, out, N_NODES);
}